// MatchPyramid_438086664310
// MI455X (gfx1250) — compile-verified
//
#include <hip/hip_runtime.h>

typedef float v2f __attribute__((ext_vector_type(2)));
typedef float v8f __attribute__((ext_vector_type(8)));

#define EPS 1e-8f

static __device__ __forceinline__ v8f wmma_f32_k4(v2f a, v2f b, v8f c) {
  // V_WMMA_F32_16X16X4_F32 : D(16x16 f32) = A(16x4 f32) * B(4x16 f32) + C
  return __builtin_amdgcn_wmma_f32_16x16x4_f32(
      /*neg_a=*/false, a, /*neg_b=*/false, b,
      /*c_mod=*/(short)0, c, /*reuse_a=*/false, /*reuse_b=*/false);
}

// ---------------------------------------------------------------------------
// Workspace layout (float offsets)
// ---------------------------------------------------------------------------
#define OFF_NA       0                        // 64*128
#define OFF_NB       (OFF_NA + 8192)          // 64*128
#define OFF_SIM      (OFF_NB + 8192)          // 64*128*128
#define OFF_P1       (OFF_SIM + 1048576)      // 64*32*62*62
#define OFF_C2       (OFF_P1 + 7872512)       // 64*64*60*60
#define OFF_P2       (OFF_C2 + 14745600)      // 64*64*30*30
#define OFF_PART     (OFF_P2 + 3686400)       // 36*64*128
#define OFF_H        (OFF_PART + 294912)      // 64*128

// ---------------------------------------------------------------------------
// Kernel 1: row norms of gathered embedding rows
// ---------------------------------------------------------------------------
__global__ void norms_kernel(const int* __restrict__ idx,
                             const float* __restrict__ emb,
                             float* __restrict__ out, int n) {
  int i = blockIdx.x * blockDim.x + threadIdx.x;
  if (i >= n) return;
  const float* row = emb + (long long)idx[i] * 300;
  float s = 0.f;
  for (int k = 0; k < 300; ++k) { float v = row[k]; s += v * v; }
  out[i] = sqrtf(s);
}

// ---------------------------------------------------------------------------
// Kernel 2: batched cosine-sim GEMM, gather fused, WMMA f32 16x16x4
//   grid (8, 64), block 256 (8 waves); wave -> one 16x16 tile of sim[b]
// ---------------------------------------------------------------------------
__global__ __launch_bounds__(256)
void sim_wmma_kernel(const int* __restrict__ ia, const int* __restrict__ ib,
                     const float* __restrict__ emb,
                     const float* __restrict__ na, const float* __restrict__ nb,
                     float* __restrict__ sim) {
  const int batch = blockIdx.y;
  const int wave  = threadIdx.x >> 5;
  const int lane  = threadIdx.x & 31;
  const int tile  = blockIdx.x * 8 + wave;   // 0..63
  const int mt = tile >> 3, nt = tile & 7;
  const int half = lane >> 4;                // 0: K=0,1  1: K=2,3
  const int l = lane & 15;

  const int rowA = ia[batch * 128 + mt * 16 + l];
  const int rowB = ib[batch * 128 + nt * 16 + l];
  const float* pa = emb + (long long)rowA * 300;
  const float* pb = emb + (long long)rowB * 300;

  v8f acc = {};
  for (int k = 0; k < 300; k += 4) {
    const int k0 = k + half * 2;
    v2f av = *(const v2f*)(pa + k0);
    v2f bv = *(const v2f*)(pb + k0);
    acc = wmma_f32_k4(av, bv, acc);
  }

  const float nbv = nb[batch * 128 + nt * 16 + l];   // this lane's column norm
  #pragma unroll
  for (int v = 0; v < 8; ++v) {
    const int m = mt * 16 + v + half * 8;            // D layout: M = v + 8*half
    const int n = nt * 16 + l;                       //           N = lane&15
    float denom = fmaxf(na[batch * 128 + m] * nbv, EPS);
    sim[((long long)batch * 128 + m) * 128 + n] = acc[v] / denom;
  }
}

// ---------------------------------------------------------------------------
// Kernel 3: conv1 (1->32, 5x5 VALID) + ReLU + 2x2 maxpool  -> p1[64][32][62][62]
//   grid (16, 32, 64), block 256
// ---------------------------------------------------------------------------
__global__ __launch_bounds__(256)
void conv1_kernel(const float* __restrict__ sim, const float* __restrict__ w,
                  const float* __restrict__ bias, float* __restrict__ p1) {
  __shared__ float wsm[25];
  const int c = blockIdx.y, b = blockIdx.z;
  if (threadIdx.x < 25) wsm[threadIdx.x] = w[c * 25 + threadIdx.x];
  __syncthreads();

  const int p = blockIdx.x * 256 + threadIdx.x;
  if (p >= 62 * 62) return;
  const int py = p / 62, px = p % 62;
  const float* s = sim + (long long)b * 128 * 128;
  const float bv = bias[c];

  float best = -3.4e38f;
  #pragma unroll
  for (int dy = 0; dy < 2; ++dy) {
    #pragma unroll
    for (int dx = 0; dx < 2; ++dx) {
      const int y0 = 2 * py + dy, x0 = 2 * px + dx;
      float acc = bv;
      #pragma unroll
      for (int r = 0; r < 5; ++r)
        #pragma unroll
        for (int q = 0; q < 5; ++q)
          acc += s[(y0 + r) * 128 + (x0 + q)] * wsm[r * 5 + q];
      best = fmaxf(best, acc);
    }
  }
  p1[(((long long)b * 32 + c) * 62 + py) * 62 + px] = fmaxf(best, 0.f);
}

// ---------------------------------------------------------------------------
// Kernel 4: conv2 (32->64, 3x3 VALID) as implicit GEMM, WMMA f32 16x16x4
//   M = 3600 pixels, N = 64 channels, K = 288
//   grid (225, 64), block 128 (4 waves = 4 N-tiles)
// ---------------------------------------------------------------------------
__global__ __launch_bounds__(128)
void conv2_wmma_kernel(const float* __restrict__ p1, const float* __restrict__ w2,
                       const float* __restrict__ b2, float* __restrict__ c2) {
  __shared__ int off[288];   // k -> input offset (ci*3844 + r*62 + s)
  for (int t = threadIdx.x; t < 288; t += 128) {
    const int ci = t / 9, rem = t % 9;
    off[t] = ci * 3844 + (rem / 3) * 62 + (rem % 3);
  }
  __syncthreads();

  const int b  = blockIdx.y;
  const int mt = blockIdx.x;            // 0..224 (225*16 = 3600 exactly)
  const int nt = threadIdx.x >> 5;      // 0..3
  const int lane = threadIdx.x & 31;
  const int half = lane >> 4, l = lane & 15;

  const int m = mt * 16 + l;            // pixel row for A loads
  const int oy = m / 60, ox = m % 60;
  const float* inb = p1 + (long long)b * 32 * 62 * 62 + oy * 62 + ox;

  const int oc = nt * 16 + l;           // channel for B loads
  const float* wrow = w2 + (long long)oc * 288;

  v8f acc = {};
  for (int k = 0; k < 288; k += 4) {
    const int k0 = k + half * 2;
    v2f av;
    av.x = inb[off[k0]];
    av.y = inb[off[k0 + 1]];
    v2f bv = *(const v2f*)(wrow + k0);
    acc = wmma_f32_k4(av, bv, acc);
  }

  const float biasv = b2[oc];
  #pragma unroll
  for (int v = 0; v < 8; ++v) {
    const int pm  = mt * 16 + v + half * 8;
    const int ocn = nt * 16 + l;
    c2[((long long)b * 64 + ocn) * 3600 + pm] = fmaxf(acc[v] + biasv, 0.f);
  }
}

// ---------------------------------------------------------------------------
// Kernel 5: 2x2 maxpool 60x60 -> 30x30   p2[64][64][30][30]
// ---------------------------------------------------------------------------
__global__ void pool2_kernel(const float* __restrict__ c2, float* __restrict__ p2) {
  int i = blockIdx.x * blockDim.x + threadIdx.x;
  if (i >= 64 * 64 * 30 * 30) return;
  const int px = i % 30, py = (i / 30) % 30, c = (i / 900) % 64, b = i / 57600;
  const float* base = c2 + ((long long)b * 64 + c) * 3600 + (2 * py) * 60 + 2 * px;
  float m0 = fmaxf(base[0], base[1]);
  float m1 = fmaxf(base[60], base[61]);
  p2[i] = fmaxf(m0, m1);
}

// ---------------------------------------------------------------------------
// Kernel 6: linear 57600 -> 128, split-K WMMA f32 16x16x4
//   grid (4 Mtiles, 8 Ntiles, 36 K-slices), block 32 (one wave)
//   Each K-slice covers 1600 K (= 400 WMMA); partials reduced deterministically.
// ---------------------------------------------------------------------------
__global__ __launch_bounds__(32)
void lin_wmma_kernel(const float* __restrict__ x, const float* __restrict__ lw,
                     float* __restrict__ partial) {
  const int mt = blockIdx.x, nt = blockIdx.y, ks = blockIdx.z;
  const int lane = threadIdx.x;
  const int half = lane >> 4, l = lane & 15;

  const float* pa = x  + (long long)(mt * 16 + l) * 57600;  // batch row
  const float* pb = lw + (long long)(nt * 16 + l) * 57600;  // hidden row

  const int kbeg = ks * 1600, kend = kbeg + 1600;
  v8f acc = {};
  for (int k = kbeg; k < kend; k += 4) {
    const int k0 = k + half * 2;
    v2f av = *(const v2f*)(pa + k0);
    v2f bv = *(const v2f*)(pb + k0);
    acc = wmma_f32_k4(av, bv, acc);
  }

  #pragma unroll
  for (int v = 0; v < 8; ++v) {
    const int m = mt * 16 + v + half * 8;
    const int n = nt * 16 + l;
    partial[(long long)ks * 8192 + m * 128 + n] = acc[v];
  }
}

// ---------------------------------------------------------------------------
// Kernel 7: deterministic split-K reduce + bias + ReLU  -> h[64][128]
// ---------------------------------------------------------------------------
__global__ void reduce_kernel(const float* __restrict__ partial,
                              const float* __restrict__ lb,
                              float* __restrict__ h) {
  int i = blockIdx.x * blockDim.x + threadIdx.x;
  if (i >= 8192) return;
  float s = lb[i % 128];
  for (int ks = 0; ks < 36; ++ks) s += partial[ks * 8192 + i];
  h[i] = fmaxf(s, 0.f);
}

// ---------------------------------------------------------------------------
// Kernel 8: prediction head 128 -> 2
// ---------------------------------------------------------------------------
__global__ void pred_kernel(const float* __restrict__ h,
                            const float* __restrict__ pw,
                            const float* __restrict__ pb,
                            float* __restrict__ out) {
  int t = threadIdx.x;             // 0..127
  if (t >= 128) return;
  const int b = t >> 1, n = t & 1;
  float s = pb[n];
  for (int j = 0; j < 128; ++j) s += h[b * 128 + j] * pw[n * 128 + j];
  out[b * 2 + n] = s;
}

// ---------------------------------------------------------------------------
extern "C" void kernel_launch(void* const* d_in, const int* in_sizes, int n_in,
                              void* d_out, int out_size, void* d_ws, size_t ws_size,
                              hipStream_t stream) {
  const int*   a    = (const int*)  d_in[0];
  const int*   bIdx = (const int*)  d_in[1];
  const float* emb  = (const float*)d_in[2];
  const float* c1w  = (const float*)d_in[3];
  const float* c1b  = (const float*)d_in[4];
  const float* c2w  = (const float*)d_in[5];
  const float* c2b  = (const float*)d_in[6];
  const float* lw   = (const float*)d_in[7];
  const float* lb   = (const float*)d_in[8];
  const float* pw   = (const float*)d_in[9];
  const float* pbv  = (const float*)d_in[10];

  float* ws   = (float*)d_ws;
  float* na   = ws + OFF_NA;
  float* nb   = ws + OFF_NB;
  float* sim  = ws + OFF_SIM;
  float* p1   = ws + OFF_P1;
  float* c2   = ws + OFF_C2;
  float* p2   = ws + OFF_P2;
  float* part = ws + OFF_PART;
  float* h    = ws + OFF_H;
  float* out  = (float*)d_out;

  norms_kernel<<<32, 256, 0, stream>>>(a, emb, na, 8192);
  norms_kernel<<<32, 256, 0, stream>>>(bIdx, emb, nb, 8192);

  sim_wmma_kernel<<<dim3(8, 64), 256, 0, stream>>>(a, bIdx, emb, na, nb, sim);

  conv1_kernel<<<dim3(16, 32, 64), 256, 0, stream>>>(sim, c1w, c1b, p1);

  conv2_wmma_kernel<<<dim3(225, 64), 128, 0, stream>>>(p1, c2w, c2b, c2);

  pool2_kernel<<<(64 * 64 * 30 * 30 + 255) / 256, 256, 0, stream>>>(c2, p2);

  lin_wmma_kernel<<<dim3(4, 8, 36), 32, 0, stream>>>(p2, lw, part);

  reduce_kernel<<<32, 256, 0, stream>>>(part, lb, h);

  pred_kernel<<<1, 128, 0, stream>>>(h, pw, pbv, out);
}